// ModelNew_19688130085433
// MI455X (gfx1250) — compile-verified
//
#include <hip/hip_runtime.h>
#include <hip/hip_bf16.h>
#include <math.h>

// ---------------------------------------------------------------------------
// NetVLAD forward for MI455X (gfx1250, wave32, WMMA f32 16x16x4 + async LDS).
//
// Shapes: x[32,2048,512] f32, clusters[512,64], clusters2[1,512,64],
//         bn_gamma[64], bn_beta[64]  ->  out[32, 512*64] f32.
// ---------------------------------------------------------------------------

typedef float v2f __attribute__((ext_vector_type(2)));
typedef float v8f __attribute__((ext_vector_type(8)));

#define D_DIM  512
#define K_DIM  64
#define B_SZ   32
#define N_SEQ  2048
#define NROW   (B_SZ * N_SEQ)        // 65536 rows of x_flat
#define NT     64                    // n-chunk staged per LDS buffer in k4

// Workspace layout (float offsets). Block 0: raw/softmax assignment (16 MB).
#define WS_STATS   (NROW * K_DIM)    // 4,194,304 floats of assignment first
#define OFF_COLSUM 0                 // [64]  sum of raw assignment per column
#define OFF_COLSQ  64                // [64]  sum of squares per column
#define OFF_SCALE  128               // [64]  BN fused scale
#define OFF_SHIFT  192               // [64]  BN fused shift
#define OFF_ASUM   256               // [32*64] per-batch softmax column sums
#define OFF_COLSS  (256 + 2048)      // [32*64] per-(b,k) sum of vlad^2 over D
#define OFF_INVC   (256 + 4096)      // [32*64] 1/max(||col||, eps)
#define OFF_INVT   (256 + 6144)      // [32]    1/max(||vlad_b||, eps)
#define STATS_ZERO_N (256 + 4096)    // zero colsum..col_ss = 4352 floats

// ---------------------------------------------------------------------------
// Kernel 0: zero the atomic accumulators in the stats scratch region.
// ---------------------------------------------------------------------------
__global__ void k0_zero(float* __restrict__ stats) {
    int i = blockIdx.x * 256 + threadIdx.x;
    if (i < STATS_ZERO_N) stats[i] = 0.0f;
}

// ---------------------------------------------------------------------------
// Kernel 1: assignment = x_flat @ clusters  (M=65536, K=512, N=64), f32 WMMA.
// Each wave owns a 16(M)x64(N) strip = 4 C tiles; block = 8 waves = 128 rows.
// Fused: per-column sum / sum-of-squares for BatchNorm (LDS, then one global
// atomic per column per block).
// ---------------------------------------------------------------------------
__global__ __launch_bounds__(256)
void k1_assign_gemm(const float* __restrict__ x,
                    const float* __restrict__ clusters,
                    float* __restrict__ assign,
                    float* __restrict__ stats) {
    __shared__ float sSum[K_DIM];
    __shared__ float sSq[K_DIM];
    const int tid = threadIdx.x;
    if (tid < K_DIM) { sSum[tid] = 0.0f; sSq[tid] = 0.0f; }
    __syncthreads();

    const int wave = tid >> 5;
    const int lane = tid & 31;
    const int lm   = lane & 15;    // row (A) / col (B,C) within tile
    const int hl   = lane >> 4;    // lane-half selects K pair / M half
    const int r0   = (blockIdx.x * 8 + wave) * 16;

    v8f acc[4] = { {}, {}, {}, {} };

    // A fragment source: lane holds x[row=r0+lm][kk + 2*hl .. +1] (8B aligned)
    const float* xrow = x + (size_t)(r0 + lm) * D_DIM;

    for (int kk = 0; kk < D_DIM; kk += 4) {
        v2f a = *(const v2f*)(xrow + kk + 2 * hl);
        // B fragments: lane holds clusters[kk+2*hl][col], clusters[kk+2*hl+1][col]
        const float* crow = clusters + (size_t)(kk + 2 * hl) * K_DIM + lm;
#pragma unroll
        for (int t = 0; t < 4; ++t) {
            v2f b;
            b.x = crow[t * 16];
            b.y = crow[t * 16 + K_DIM];
            acc[t] = __builtin_amdgcn_wmma_f32_16x16x4_f32(
                false, a, false, b, (short)0, acc[t], false, false);
        }
    }

    // Write raw assignment + fold column statistics.
#pragma unroll
    for (int t = 0; t < 4; ++t) {
        float s = 0.0f, q = 0.0f;
        const int col = t * 16 + lm;
#pragma unroll
        for (int j = 0; j < 8; ++j) {
            const int row = r0 + hl * 8 + j;          // C layout: M = vgpr + 8*half
            const float v = acc[t][j];
            assign[(size_t)row * K_DIM + col] = v;
            s += v;
            q += v * v;
        }
        s += __shfl_xor(s, 16);                       // combine M halves (same col)
        q += __shfl_xor(q, 16);
        if (lane < 16) {
            atomicAdd(&sSum[col], s);
            atomicAdd(&sSq[col], q);
        }
    }
    __syncthreads();
    if (tid < K_DIM) {
        atomicAdd(&stats[OFF_COLSUM + tid], sSum[tid]);
        atomicAdd(&stats[OFF_COLSQ + tid], sSq[tid]);
    }
}

// ---------------------------------------------------------------------------
// Kernel 2: fold BatchNorm into per-column scale/shift.
// ---------------------------------------------------------------------------
__global__ void k2_bn_params(const float* __restrict__ gamma,
                             const float* __restrict__ beta,
                             float* __restrict__ stats) {
    const int k = threadIdx.x;                        // 64 threads
    const float invn = 1.0f / (float)NROW;
    const float mean = stats[OFF_COLSUM + k] * invn;
    const float var  = stats[OFF_COLSQ + k] * invn - mean * mean;
    const float sc   = gamma[k] * rsqrtf(var + 1e-5f);
    stats[OFF_SCALE + k] = sc;
    stats[OFF_SHIFT + k] = beta[k] - mean * sc;
}

// ---------------------------------------------------------------------------
// Kernel 3: per-row softmax over K=64 (one wave per row, 2 cols/lane) and
// fused per-batch a_sum accumulation. 8 rows per block never straddle a
// batch boundary (2048 % 8 == 0).
// ---------------------------------------------------------------------------
__global__ __launch_bounds__(256)
void k3_softmax(float* __restrict__ assign, float* __restrict__ stats) {
    __shared__ float sA[K_DIM];
    const int tid = threadIdx.x;
    if (tid < K_DIM) sA[tid] = 0.0f;
    __syncthreads();

    const int wave = tid >> 5;
    const int lane = tid & 31;
    const int row  = blockIdx.x * 8 + wave;
    float* ar = assign + (size_t)row * K_DIM;

    const float y0 = ar[lane]      * stats[OFF_SCALE + lane]      + stats[OFF_SHIFT + lane];
    const float y1 = ar[lane + 32] * stats[OFF_SCALE + lane + 32] + stats[OFF_SHIFT + lane + 32];

    float m = fmaxf(y0, y1);
#pragma unroll
    for (int o = 16; o > 0; o >>= 1) m = fmaxf(m, __shfl_xor(m, o));
    float e0 = __expf(y0 - m), e1 = __expf(y1 - m);
    float s = e0 + e1;
#pragma unroll
    for (int o = 16; o > 0; o >>= 1) s += __shfl_xor(s, o);
    const float inv = 1.0f / s;
    e0 *= inv; e1 *= inv;
    ar[lane]      = e0;
    ar[lane + 32] = e1;

    atomicAdd(&sA[lane],      e0);
    atomicAdd(&sA[lane + 32], e1);
    __syncthreads();
    if (tid < K_DIM) {
        const int b = blockIdx.x >> 8;                // (blockIdx*8) / 2048
        atomicAdd(&stats[OFF_ASUM + b * K_DIM + tid], sA[tid]);
    }
}

// ---------------------------------------------------------------------------
// CDNA5 async copy: 16B per lane, global -> LDS, tracked by ASYNCcnt.
// No portable clang builtin for this op -> inline asm (CDNA5 ISA).
// ---------------------------------------------------------------------------
__device__ __forceinline__ void async_b128(unsigned lds_off, const void* gaddr) {
    asm volatile("global_load_async_to_lds_b128 %0, %1, off"
                 :: "v"(lds_off), "v"(gaddr) : "memory");
}
__device__ __forceinline__ void wait_async0() {
    asm volatile("s_wait_asynccnt 0" ::: "memory");
}

// ---------------------------------------------------------------------------
// Kernel 4: vlad[b,d,k] = sum_n assign[b,n,k]*x[b,n,d] - a_sum[b,k]*c2[d,k].
// Per batch: M=D=512 rows, N=K=64 cols, inner=2048 (f32 WMMA 16x16x4).
// Block = 8 waves = 32 d-rows x all 64 k for one batch. Tiles of 64 n-rows
// are double-buffered in LDS via GLOBAL_LOAD_ASYNC_TO_LDS_B128 (fully
// coalesced, de-duplicated across waves, fetch overlapped with WMMA).
// Raw vlad goes straight into d_out ([B,D,K] flat == output layout); fused
// per-(b,k) sum of squares for intra-normalization.
// ---------------------------------------------------------------------------
__global__ __launch_bounds__(256)
void k4_vlad_gemm(const float* __restrict__ x,
                  const float* __restrict__ assign,
                  const float* __restrict__ clusters2,
                  float* __restrict__ out,
                  float* __restrict__ stats) {
    __shared__ float sx[2][NT * 32];   // x tile:      [n_local][d_local]  8KB/buf
    __shared__ float sa[2][NT * 64];   // assign tile: [n_local][k]       16KB/buf

    const int tid   = threadIdx.x;
    const int wave  = tid >> 5;
    const int lane  = tid & 31;
    const int lm    = lane & 15;
    const int hl    = lane >> 4;
    const int b     = blockIdx.y;
    const int dblk  = blockIdx.x * 32;
    const int dloc0 = (wave & 1) * 16;                  // d offset inside block tile
    const int k0    = (wave >> 1) * 16;                 // k-tile base

    const float* xb = x      + (size_t)b * N_SEQ * D_DIM + dblk;
    const float* ab = assign + (size_t)b * N_SEQ * K_DIM;

    // LDS byte offsets (generic LDS pointer: low 32 bits == LDS offset)
    const unsigned sx_base = (unsigned)(uintptr_t)&sx[0][0];
    const unsigned sa_base = (unsigned)(uintptr_t)&sa[0][0];

    auto issue = [&](int buf, int n0) {
        const unsigned sxb = sx_base + (unsigned)buf * (NT * 32 * 4);
        const unsigned sab = sa_base + (unsigned)buf * (NT * 64 * 4);
        // x tile: 64 rows x 128B; 512 b128 transfers, 2 per thread.
        // lanes 0..7 of a wave cover one full contiguous 128B row.
#pragma unroll
        for (int p = 0; p < 2; ++p) {
            const int t = tid + p * 256;
            const int row = t >> 3, seg = t & 7;
            async_b128(sxb + (unsigned)(row * 32 + seg * 4) * 4,
                       xb + (size_t)(n0 + row) * D_DIM + seg * 4);
        }
        // assign tile: 64 rows x 256B; 1024 b128 transfers, 4 per thread.
        // lanes 0..15 of a wave cover one full contiguous 256B row.
#pragma unroll
        for (int p = 0; p < 4; ++p) {
            const int t = tid + p * 256;
            const int row = t >> 4, seg = t & 15;
            async_b128(sab + (unsigned)(row * 64 + seg * 4) * 4,
                       ab + (size_t)(n0 + row) * K_DIM + seg * 4);
        }
    };

    v8f c = {};
    issue(0, 0);
    for (int chunk = 0; chunk < N_SEQ / NT; ++chunk) {
        const int buf = chunk & 1;
        wait_async0();                 // my chunk-i copies have landed in LDS
        __syncthreads();               // everyone's copies have landed
        if (chunk + 1 < N_SEQ / NT)    // prefetch next chunk into other buffer
            issue(buf ^ 1, (chunk + 1) * NT);

        const float* tx = &sx[buf][0];
        const float* ta = &sa[buf][0];
#pragma unroll
        for (int kk = 0; kk < NT; kk += 4) {
            const int na = kk + 2 * hl;
            v2f a, bb;
            a.x  = tx[na * 32 + dloc0 + lm];            // A = x^T fragment
            a.y  = tx[(na + 1) * 32 + dloc0 + lm];
            bb.x = ta[na * 64 + k0 + lm];               // B = assignment fragment
            bb.y = ta[(na + 1) * 64 + k0 + lm];
            c = __builtin_amdgcn_wmma_f32_16x16x4_f32(
                false, a, false, bb, (short)0, c, false, false);
        }
        __syncthreads();               // done reading buf before it is refilled
    }

    const int k    = k0 + lm;
    const int dt0  = dblk + dloc0;
    const float asum = stats[OFF_ASUM + b * K_DIM + k];
    float* ob = out + (size_t)b * D_DIM * K_DIM;
    float ss = 0.0f;
#pragma unroll
    for (int j = 0; j < 8; ++j) {
        const int d = dt0 + hl * 8 + j;                 // C layout: M = vgpr + 8*half
        const float v = c[j] - asum * clusters2[(size_t)d * K_DIM + k];
        ob[(size_t)d * K_DIM + k] = v;
        ss += v * v;
    }
    ss += __shfl_xor(ss, 16);                           // both halves share column k
    if (lane < 16) atomicAdd(&stats[OFF_COLSS + b * K_DIM + k], ss);
}

// ---------------------------------------------------------------------------
// Kernel 5: per-(b,k) inverse intra-norm and per-batch inverse global norm.
// ---------------------------------------------------------------------------
__global__ void k5_norms(float* __restrict__ stats) {
    __shared__ float red[K_DIM];
    const int b = blockIdx.x;
    const int k = threadIdx.x;                           // 64 threads
    const float ss  = stats[OFF_COLSS + b * K_DIM + k];
    const float ic  = 1.0f / fmaxf(sqrtf(ss), 1e-12f);
    stats[OFF_INVC + b * K_DIM + k] = ic;
    red[k] = ss * ic * ic;                               // ||col||^2 after intra-norm
    __syncthreads();
#pragma unroll
    for (int o = 32; o > 0; o >>= 1) {
        if (k < o) red[k] += red[k + o];
        __syncthreads();
    }
    if (k == 0) stats[OFF_INVT + b] = 1.0f / fmaxf(sqrtf(red[0]), 1e-12f);
}

// ---------------------------------------------------------------------------
// Kernel 6: out[b, d*64+k] *= inv_col[b,k] * inv_tot[b]   (in place).
// ---------------------------------------------------------------------------
__global__ __launch_bounds__(256)
void k6_finalize(float* __restrict__ out, const float* __restrict__ stats) {
    const int idx = blockIdx.x * 256 + threadIdx.x;      // 1,048,576 elements
    const int b = idx >> 15;                             // / (512*64)
    const int k = idx & 63;
    out[idx] *= stats[OFF_INVC + b * K_DIM + k] * stats[OFF_INVT + b];
}

// ---------------------------------------------------------------------------
extern "C" void kernel_launch(void* const* d_in, const int* in_sizes, int n_in,
                              void* d_out, int out_size, void* d_ws, size_t ws_size,
                              hipStream_t stream) {
    const float* x         = (const float*)d_in[0];
    const float* clusters  = (const float*)d_in[1];
    const float* clusters2 = (const float*)d_in[2];
    const float* bn_gamma  = (const float*)d_in[3];
    const float* bn_beta   = (const float*)d_in[4];
    float* out = (float*)d_out;

    float* ws     = (float*)d_ws;
    float* assign = ws;                 // [65536, 64]
    float* stats  = ws + WS_STATS;      // small accumulators

    k0_zero<<<(STATS_ZERO_N + 255) / 256, 256, 0, stream>>>(stats);
    k1_assign_gemm<<<NROW / 128, 256, 0, stream>>>(x, clusters, assign, stats);
    k2_bn_params<<<1, K_DIM, 0, stream>>>(bn_gamma, bn_beta, stats);
    k3_softmax<<<NROW / 8, 256, 0, stream>>>(assign, stats);
    k4_vlad_gemm<<<dim3(D_DIM / 32, B_SZ), 256, 0, stream>>>(x, assign, clusters2, out, stats);
    k5_norms<<<B_SZ, K_DIM, 0, stream>>>(stats);
    k6_finalize<<<(B_SZ * D_DIM * K_DIM) / 256, 256, 0, stream>>>(out, stats);
}